// GCN_Net_377957122119
// MI455X (gfx1250) — compile-verified
//
#include <hip/hip_runtime.h>

typedef __attribute__((ext_vector_type(2))) float v2f;
typedef __attribute__((ext_vector_type(8))) float v8f;

#define DIM  128
#define LDSP 132   // LDS pitch: 132 mod 64 == 4 banks -> conflict-free b64 reads

// ---------------- degree / normalization ----------------

__global__ __launch_bounds__(256) void k_deg_init(float* deg, int n) {
  int i = blockIdx.x * 256 + threadIdx.x;
  if (i < n) deg[i] = 1.0f;                       // self loop
}

__global__ __launch_bounds__(256) void k_deg_edges(const int* __restrict__ dst,
                                                   float* deg, int ne) {
  int e = blockIdx.x * 256 + threadIdx.x;
  if (e < ne) atomicAdd(&deg[dst[e]], 1.0f);
}

__global__ __launch_bounds__(256) void k_rsqrt(float* deg, int n) {
  int i = blockIdx.x * 256 + threadIdx.x;
  if (i < n) deg[i] = rsqrtf(deg[i]);             // dinv = deg^-1/2
}

// ---------------- GEMM: Out = (relu?)A[nrows,128] @ W[128,128] ----------------
// One wave -> 16x128 output slab: 8 N-tiles, K=128 in steps of 4 via
// V_WMMA_F32_16X16X4_F32 (f32 in / f32 accum, matches reference numerics).
// RELU is a compile-time template arg so the non-relu instantiation carries
// zero select/max overhead in the inner loop.

template <int RELU>
__global__ __launch_bounds__(256) void k_gemm_wmma(const float* __restrict__ A,
                                                   const float* __restrict__ W,
                                                   float* __restrict__ Out,
                                                   int nrows) {
  __shared__ float lw[DIM * LDSP];                // W transposed: lw[n*LDSP + k]
  const int tid = threadIdx.x;
  for (int idx = tid; idx < DIM * DIM; idx += 256) {
    int k = idx >> 7, n = idx & (DIM - 1);
    lw[n * LDSP + k] = W[idx];
  }
  __syncthreads();

  const int lane  = tid & 31;
  const int lrow  = lane & 15;          // M index for A, N index for B/D
  const int khalf = (lane >> 4) << 1;   // 0 or 2 (upper half-wave = K+2)
  const size_t m0 = ((size_t)blockIdx.x * 8 + (tid >> 5)) * 16;
  if (m0 >= (size_t)nrows) return;

  v8f acc[8] = {};
  const float* arow = A + (m0 + lrow) * DIM;

  for (int k0 = 0; k0 < DIM; k0 += 4) {
    const int ka = k0 + khalf;
    // A 16x4 f32 fragment: lane(lrow) holds A[M=lrow][ka], A[M=lrow][ka+1]
    v2f a = *(const v2f*)(arow + ka);
    if (RELU) { a.x = fmaxf(a.x, 0.0f); a.y = fmaxf(a.y, 0.0f); }
#pragma unroll
    for (int t = 0; t < 8; ++t) {
      // B 4x16 f32 fragment: lane(lrow) holds W[ka][n], W[ka+1][n], n = t*16+lrow
      v2f b = *(const v2f*)(&lw[(t * 16 + lrow) * LDSP + ka]);
      acc[t] = __builtin_amdgcn_wmma_f32_16x16x4_f32(
          false, a, false, b, (short)0, acc[t], false, false);
    }
  }

  // D 16x16 f32 layout: VGPR r, lanes 0-15 -> M=r, lanes 16-31 -> M=r+8; N=lrow
  const int rbase = (lane >> 4) << 3;   // 0 or 8
#pragma unroll
  for (int t = 0; t < 8; ++t) {
    float* op = Out + (m0 + rbase) * DIM + t * 16 + lrow;
#pragma unroll
    for (int r = 0; r < 8; ++r) op[(size_t)r * DIM] = acc[t][r];
  }
}

// ---------------- aggregation ----------------
// out = dinv[i]^2 * h[i] + bias   (self-loop term + bias, init pass)
__global__ __launch_bounds__(256) void k_agg_init(const float* __restrict__ h,
                                                  const float* __restrict__ dinv,
                                                  const float* __restrict__ bias,
                                                  float* __restrict__ out,
                                                  int nnodes) {
  size_t idx = (size_t)blockIdx.x * 256 + threadIdx.x;
  size_t total = (size_t)nnodes * DIM;
  if (idx >= total) return;
  int node = (int)(idx >> 7);
  int c    = (int)(idx & (DIM - 1));
  float di = dinv[node];
  out[idx] = di * di * h[idx] + bias[c];
}

// One wave per edge: lane covers 4 channels (float4 gather, 4 f32 atomics)
__global__ __launch_bounds__(256) void k_agg_edges(const int* __restrict__ src,
                                                   const int* __restrict__ dst,
                                                   const float* __restrict__ dinv,
                                                   const float* __restrict__ h,
                                                   float* __restrict__ out,
                                                   int ne) {
  int e    = (int)(((size_t)blockIdx.x * 256 + threadIdx.x) >> 5);
  int lane = threadIdx.x & 31;
  if (e >= ne) return;
  int s = src[e], d = dst[e];
  float norm = dinv[s] * dinv[d];
  const float4* hp = (const float4*)(h + (size_t)s * DIM);
  float4 v = hp[lane];
  float* op = out + (size_t)d * DIM + lane * 4;
  atomicAdd(op + 0, v.x * norm);
  atomicAdd(op + 1, v.y * norm);
  atomicAdd(op + 2, v.z * norm);
  atomicAdd(op + 3, v.w * norm);
}

// ---------------- ragged per-graph mean pool ----------------
__global__ __launch_bounds__(DIM) void k_pool(const float* __restrict__ h2,
                                              const int* __restrict__ scope,
                                              float* __restrict__ out,
                                              int ngraphs) {
  int g = blockIdx.x;
  int c = threadIdx.x;
  if (g >= ngraphs) return;
  size_t start = 0;
  for (int i = 0; i < g; ++i) start += (size_t)scope[i];
  int cnt = scope[g];
  float s = 0.0f;
  for (int j = 0; j < cnt; ++j) s += h2[(start + j) * DIM + c];
  out[(size_t)g * DIM + c] = s / (float)cnt;
}

// ---------------- launch ----------------
extern "C" void kernel_launch(void* const* d_in, const int* in_sizes, int n_in,
                              void* d_out, int out_size, void* d_ws, size_t ws_size,
                              hipStream_t stream) {
  (void)n_in; (void)out_size; (void)ws_size;
  const float* x     = (const float*)d_in[0];
  const int*   ei    = (const int*)d_in[1];
  const int*   scope = (const int*)d_in[2];
  const float* W1    = (const float*)d_in[3];
  const float* b1    = (const float*)d_in[4];
  const float* W2    = (const float*)d_in[5];
  const float* b2    = (const float*)d_in[6];
  float*       out   = (float*)d_out;

  const int n_nodes  = in_sizes[0] / DIM;
  const int n_edges  = in_sizes[1] / 2;
  const int n_graphs = in_sizes[2];
  const int* src = ei;
  const int* dst = ei + n_edges;

  float* dinv = (float*)d_ws;                       // [n_nodes]
  float* bufA = dinv + n_nodes;                     // [n_nodes*128]
  float* bufB = bufA + (size_t)n_nodes * DIM;       // [n_nodes*128]

  const int gb_nodes  = (n_nodes + 255) / 256;
  const int gb_edges  = (n_edges + 255) / 256;
  const size_t total  = (size_t)n_nodes * DIM;
  const int gb_elems  = (int)((total + 255) / 256);
  const int mtiles    = (n_nodes + 15) / 16;
  const int gb_gemm   = (mtiles + 7) / 8;           // 8 waves (M-tiles) per block
  const int gb_wedges = (n_edges + 7) / 8;          // 8 edges (waves) per block

  // normalization: deg = 1 + in-degree; dinv = rsqrt(deg)
  k_deg_init <<<gb_nodes, 256, 0, stream>>>(dinv, n_nodes);
  k_deg_edges<<<gb_edges, 256, 0, stream>>>(dst, dinv, n_edges);
  k_rsqrt    <<<gb_nodes, 256, 0, stream>>>(dinv, n_nodes);

  // layer 1: h1 = x@W1 ; agg1 = scatter(norm*h1) + self + b1
  k_gemm_wmma<0><<<gb_gemm,   256, 0, stream>>>(x, W1, bufA, n_nodes);
  k_agg_init    <<<gb_elems,  256, 0, stream>>>(bufA, dinv, b1, bufB, n_nodes);
  k_agg_edges   <<<gb_wedges, 256, 0, stream>>>(src, dst, dinv, bufA, bufB, n_edges);

  // layer 2: h2 = relu(agg1)@W2 ; agg2 = scatter(norm*h2) + self + b2
  k_gemm_wmma<1><<<gb_gemm,   256, 0, stream>>>(bufB, W2, bufA, n_nodes);
  k_agg_init    <<<gb_elems,  256, 0, stream>>>(bufA, dinv, b2, bufB, n_nodes);
  k_agg_edges   <<<gb_wedges, 256, 0, stream>>>(src, dst, dinv, bufA, bufB, n_edges);

  // ragged mean pool
  k_pool<<<n_graphs, DIM, 0, stream>>>(bufB, scope, out, n_graphs);
}